// HYPERGC_63788854280912
// MI455X (gfx1250) — compile-verified
//
#include <hip/hip_runtime.h>
#include <math.h>

typedef __attribute__((ext_vector_type(2))) float v2f;
typedef __attribute__((ext_vector_type(8))) float v8f;
typedef unsigned int u32x4 __attribute__((ext_vector_type(4)));
typedef int          i32x4 __attribute__((ext_vector_type(4)));
typedef int          i32x8 __attribute__((ext_vector_type(8)));

#define N_    64
#define C_    64
#define T_    256
#define V_    25
#define NS_   8
#define HD_   16
#define CI_   8          // C_/NS_
#define KSEL  9
#define YSIZE (N_*C_*T_*V_)     // 26,214,400
#define TCHUNK 16
#define COLS  (TCHUNK*V_)       // 400

// ---------------------------------------------------------------------------
// Kernel 1: t_x[n,c,v] = mean over T of x[n,c,t,v].  One block per (n,c).
// ---------------------------------------------------------------------------
__global__ __launch_bounds__(256)
void hg_tmean(const float* __restrict__ x, float* __restrict__ tx)
{
    __shared__ float part[200];
    const int tid = threadIdx.x;
    const long base = (long)blockIdx.x * (T_ * V_);
    if (tid < 200) {
        const int v = tid % 25, slice = tid / 25;   // 8 slices of 32 t's
        float s = 0.f;
        #pragma unroll 4
        for (int k = 0; k < 32; ++k)
            s += x[base + (long)(slice * 32 + k) * V_ + v];
        part[tid] = s;
    }
    __syncthreads();
    if (tid < 25) {
        float s = 0.f;
        #pragma unroll
        for (int sl = 0; sl < 8; ++sl) s += part[sl * 25 + tid];
        tx[(long)blockIdx.x * V_ + tid] = s * (1.0f / 256.0f);
    }
}

// ---------------------------------------------------------------------------
// Kernel 2: per-sample semantic hypergraph -> A_fused[n] (25x25).
// One block per n.  All tensors tiny; lives in dynamic LDS.
// ---------------------------------------------------------------------------
__global__ __launch_bounds__(256)
void hg_attn(const float* __restrict__ tx, const float* __restrict__ PA,
             const float* __restrict__ EI,
             const float* __restrict__ Wq, const float* __restrict__ bq,
             const float* __restrict__ Wk, const float* __restrict__ bk,
             const float* __restrict__ Ww1, const float* __restrict__ bw1,
             const float* __restrict__ Ww2, const float* __restrict__ bw2,
             const float* __restrict__ conf, const float* __restrict__ alpha,
             float* __restrict__ afused)
{
    extern __shared__ float sm[];
    float* txs  = sm;            // 1600   t_x[n] as [c*25+v]
    float* Qb   = txs + 1600;    // 3200   [(g*25+v)*16+d]
    float* Kb   = Qb + 3200;     // 3200
    float* hb   = Kb + 3200;     // 3200   [(g*16+d)*25+v]
    float* Ah   = hb + 3200;     // 5000   [(g*25+u)*25+v] (becomes H in place)
    float* wb   = Ah + 5000;     // 200    [o*25+v]
    float* eb   = wb + 200;      // 8
    float* om   = eb + 8;        // 8
    float* Hs   = om + 8;        // 625
    float* rH   = Hs + 625;      // 25
    float* rP   = rH + 25;       // 25

    const int tid = threadIdx.x;
    const int n = blockIdx.x;

    for (int i = tid; i < 1600; i += 256) txs[i] = tx[(long)n * 1600 + i];
    __syncthreads();

    // grouped 1x1 convs: Q, K, and confidence branch h (LeakyReLU)
    for (int idx = tid; idx < NS_ * V_ * HD_; idx += 256) {
        const int g = idx / (V_ * HD_);
        const int rem = idx - g * (V_ * HD_);
        const int v = rem / HD_, d = rem - v * HD_;
        const int ch = g * HD_ + d;
        float q = bq[ch], kk = bk[ch], hv = bw1[ch];
        #pragma unroll
        for (int i = 0; i < CI_; ++i) {
            const float xv = txs[(g * CI_ + i) * V_ + v];
            q  += Wq [ch * CI_ + i] * xv;
            kk += Wk [ch * CI_ + i] * xv;
            hv += Ww1[ch * CI_ + i] * xv;
        }
        Qb[(g * V_ + v) * HD_ + d] = q;
        Kb[(g * V_ + v) * HD_ + d] = kk;
        hb[ch * V_ + v] = (hv > 0.f) ? hv : 0.01f * hv;   // LeakyReLU(0.01)
    }
    __syncthreads();

    // A_h[g,u,v] = scale * <Q[g,u,:], K[g,v,:]>
    for (int idx = tid; idx < NS_ * V_ * V_; idx += 256) {
        const int g = idx / (V_ * V_);
        const int r = idx - g * (V_ * V_);
        const int u = r / V_, v = r - u * V_;
        float s = 0.f;
        #pragma unroll
        for (int d = 0; d < HD_; ++d)
            s += Qb[(g * V_ + u) * HD_ + d] * Kb[(g * V_ + v) * HD_ + d];
        Ah[idx] = s * 0.25f;   // HD^-0.5
    }
    __syncthreads();

    // per-row top-9 threshold + masked softmax (in place)
    if (tid < NS_ * V_) {
        float* row = &Ah[tid * V_];
        float r[V_];
        #pragma unroll
        for (int v = 0; v < V_; ++v) r[v] = row[v];
        float prev = 3.4e38f, thr = -3.4e38f;
        #pragma unroll
        for (int it = 0; it < KSEL; ++it) {
            float m = -3.4e38f;
            #pragma unroll
            for (int v = 0; v < V_; ++v)
                if (r[v] < prev && r[v] > m) m = r[v];
            prev = m;
        }
        thr = prev;                       // 9th-largest value
        float mx = -3.4e38f;
        #pragma unroll
        for (int v = 0; v < V_; ++v) if (r[v] > mx) mx = r[v];
        float sum = 0.f;
        #pragma unroll
        for (int v = 0; v < V_; ++v)
            if (r[v] >= thr) sum += __expf(r[v] - mx);
        const float inv = 1.0f / sum;
        #pragma unroll
        for (int v = 0; v < V_; ++v)
            row[v] = (r[v] >= thr) ? __expf(r[v] - mx) * inv : 0.f;
    }

    // confidence: w[o,v] = tanh(Ww2 . h + bw2)
    for (int idx = tid; idx < NS_ * V_; idx += 256) {
        const int o = idx / V_, v = idx - o * V_;
        float s = bw2[o];
        #pragma unroll 8
        for (int c = 0; c < NS_ * HD_; ++c)
            s += Ww2[o * (NS_ * HD_) + c] * hb[c * V_ + v];
        wb[idx] = tanhf(s);
    }
    __syncthreads();

    if (tid < NS_) {
        float s = 0.f;
        #pragma unroll
        for (int v = 0; v < V_; ++v) s += wb[tid * V_ + v];
        eb[tid] = conf[tid] + s * (1.0f / V_);
    }
    __syncthreads();
    if (tid < NS_) {                      // softmax over heads (redundant per thread)
        float m = -3.4e38f;
        #pragma unroll
        for (int g = 0; g < NS_; ++g) m = fmaxf(m, eb[g]);
        float sum = 0.f;
        #pragma unroll
        for (int g = 0; g < NS_; ++g) sum += __expf(eb[g] - m);
        om[tid] = __expf(eb[tid] - m) / sum;
    }
    __syncthreads();

    // H_sem = sum_g omega[g] * H[g]
    for (int idx = tid; idx < V_ * V_; idx += 256) {
        float s = 0.f;
        #pragma unroll
        for (int g = 0; g < NS_; ++g) s += Ah[g * (V_ * V_) + idx] * om[g];
        Hs[idx] = s;
    }
    __syncthreads();

    if (tid < V_) {                       // L1 row norms for A_sem and A_learn
        float sh = 0.f, sp = 0.f;
        #pragma unroll
        for (int v = 0; v < V_; ++v) {
            sh += fabsf(Hs[tid * V_ + v]);
            sp += fabsf(EI[tid * V_ + v] * PA[tid * V_ + v]);
        }
        rH[tid] = 1.0f / (sh + 1e-8f);
        rP[tid] = 1.0f / (sp + 1e-8f);
    }
    __syncthreads();

    const float al = fmaxf(alpha[0], 0.f);
    for (int idx = tid; idx < V_ * V_; idx += 256) {
        const int u = idx / V_;
        afused[(long)n * (V_ * V_) + idx] =
            EI[idx] * PA[idx] * rP[u] + al * Hs[idx] * rH[u];
    }
}

// ---------------------------------------------------------------------------
// TDM: DMA a 2D tile (64 rows x 400 floats, row stride 6400 floats) from
// global memory into LDS.  D# bit layout per CDNA5 ISA ch.8 (Tensor DMA
// Descriptor): group0 = {count, lds_addr, global_addr, type=2},
// group1 = {data_size=4B, tensor/tile dims, dim0 stride}.  2D tensor ->
// groups 2/3 zero.  Issued by one wave; EXEC-independent; tracked on
// TENSORcnt.
// ---------------------------------------------------------------------------
__device__ __forceinline__
void tdm_load_tile_64x400(const float* gsrc, float* lds_dst)
{
    const unsigned long long ga = (unsigned long long)(uintptr_t)gsrc;
    const unsigned lds = (unsigned)(uintptr_t)lds_dst;   // LDS aperture: low 32 bits = LDS byte offset

    u32x4 g0 = (u32x4)0;
    g0[0] = 1u;                                          // count=1 (valid), load, no gather
    g0[1] = lds;                                         // lds_addr [63:32]
    g0[2] = (unsigned)(ga & 0xffffffffu);                // global_addr [95:64]
    g0[3] = (unsigned)((ga >> 32) & 0x01ffffffu)         // global_addr [120:96]
          | (2u << 30);                                  // type=2 ("image") [127:126]

    i32x8 g1 = (i32x8)0;
    g1[0] = (int)(2u << 16);            // workgroup_mask=0 | data_size=2 (4 bytes)
    g1[1] = (int)(400u << 16);          // tensor_dim0[15:0] = 400   (bits 79:48)
    g1[2] = (int)(64u  << 16);          // tensor_dim0 hi=0 | tensor_dim1[15:0]=64
    g1[3] = (int)(400u << 16);          // tensor_dim1 hi=0 | tile_dim0=400
    g1[4] = (int)(64u);                 // tile_dim1=64 | tile_dim2=0
    g1[5] = (int)(6400u);               // tensor_dim0_stride = 6400 (elements)
    g1[6] = 0;                          // stride hi | tensor_dim1_stride lo
    g1[7] = 0;

    i32x4 g2 = (i32x4)0;                // 2D: groups 2/3 unused
    i32x4 g3 = (i32x4)0;

#if __has_include(<hip/amd_detail/amd_gfx1250_TDM.h>)
    i32x8 g4 = (i32x8)0;                // therock/clang-23: 6-arg form
    __builtin_amdgcn_tensor_load_to_lds(g0, g1, g2, g3, g4, 0);
#else
    __builtin_amdgcn_tensor_load_to_lds(g0, g1, g2, g3, 0);   // ROCm 7.2 / clang-22
#endif
}

// ---------------------------------------------------------------------------
// Kernel 3: y = ReLU( BN( Wd * (x @_v A_fused^T) + bd*rowsum(A) ) + x )
// One block per (n, 16-t chunk).  TDM stages x-tile; WMMA f32 16x16x4.
// ---------------------------------------------------------------------------
__global__ __launch_bounds__(256)
void hg_fused(const float* __restrict__ x,  const float* __restrict__ Wd,
              const float* __restrict__ bd, const float* __restrict__ gamma,
              const float* __restrict__ beta, const float* __restrict__ mean,
              const float* __restrict__ var,  const float* __restrict__ afused,
              float* __restrict__ y)
{
    extern __shared__ float sm[];
    float* xs  = sm;               // 25600  x[c][col], col = tloc*25 + v
    float* zs  = xs + 25600;       // 25600  z[c][col], col = tloc*25 + u
    float* wd  = zs + 25600;       // 4096   Wd row-major [o][i]
    float* af  = wd + 4096;        // 650    A_fused, stride 26
    float* rsA = af + 650;         // 25     rowsum(A)
    float* bdl = rsA + 25;         // 64
    float* bsc = bdl + 64;         // 64     BN scale
    float* bsh = bsc + 64;         // 64     BN shift

    const int tid = threadIdx.x;
    const int n  = blockIdx.x >> 4;
    const int t0 = (blockIdx.x & 15) * TCHUNK;
    const long base = (long)n * (C_ * T_ * V_) + (long)t0 * V_;

    // wave 0 kicks off the async DMA of the whole x-tile (100 KB) into LDS
    const bool wave0 = (__builtin_amdgcn_readfirstlane((int)threadIdx.x) < 32);
    if (wave0)
        tdm_load_tile_64x400(x + base, xs);

    // meanwhile: small parameter loads on the other waves
    for (int i = tid; i < 4096; i += 256) wd[i] = Wd[i];
    for (int i = tid; i < 625; i += 256)
        af[(i / 25) * 26 + (i % 25)] = afused[(long)n * 625 + i];
    if (tid < C_) {
        bdl[tid] = bd[tid];
        const float sc = gamma[tid] * rsqrtf(var[tid] + 1e-5f);
        bsc[tid] = sc;
        bsh[tid] = beta[tid] - mean[tid] * sc;
    }

    if (wave0)
        __builtin_amdgcn_s_wait_tensorcnt(0);   // DMA complete before barrier release
    __syncthreads();

    if (tid < V_) {
        float s = 0.f;
        #pragma unroll
        for (int v = 0; v < V_; ++v) s += af[tid * 26 + v];
        rsA[tid] = s;
    }
    __syncthreads();

    // stage 1 (VALU): z[c, t, u] = sum_v x[c,t,v] * A[u,v]
    for (int p = tid; p < C_ * TCHUNK; p += 256) {
        const int c = p / TCHUNK, tl = p - c * TCHUNK;
        const int rb = c * COLS + tl * V_;
        float xr[V_];
        #pragma unroll
        for (int v = 0; v < V_; ++v) xr[v] = xs[rb + v];
        #pragma unroll
        for (int u = 0; u < V_; ++u) {
            float acc = 0.f;
            #pragma unroll
            for (int v = 0; v < V_; ++v) acc += xr[v] * af[u * 26 + v];
            zs[rb + u] = acc;
        }
    }
    __syncthreads();

    // stage 2 (WMMA): y[c, col] = Wd(64x64) * z(64 x 400), fused epilogue
    const int wave = tid >> 5, lane = tid & 31;
    const int half = lane >> 4, l = lane & 15;
    for (int tile = wave; tile < 4 * (COLS / 16); tile += 8) {
        const int c0 = (tile / (COLS / 16)) * 16;
        const int n0 = (tile % (COLS / 16)) * 16;
        v8f acc = {};
        #pragma unroll
        for (int k = 0; k < C_; k += 4) {
            v2f a, b;
            // A 16x4 f32: VGPR0 holds K = k+2*half, VGPR1 holds K = k+1+2*half
            a.x = wd[(c0 + l) * C_ + k + 2 * half];
            a.y = wd[(c0 + l) * C_ + k + 1 + 2 * half];
            // B 4x16 f32: rows striped across lanes within a VGPR
            b.x = zs[(k + 2 * half) * COLS + n0 + l];
            b.y = zs[(k + 1 + 2 * half) * COLS + n0 + l];
            acc = __builtin_amdgcn_wmma_f32_16x16x4_f32(
                      false, a, false, b, (short)0, acc, false, false);
        }
        const int col = n0 + l;
        const int u = col % V_;
        const float rs = rsA[u];
        #pragma unroll
        for (int r = 0; r < 8; ++r) {
            const int c = c0 + r + 8 * half;     // D layout: VGPR r -> M = r (+8 hi half)
            float v = acc[r] + bdl[c] * rs;      // bd folded through aggregation
            v = v * bsc[c] + bsh[c];             // BatchNorm (inference)
            v += xs[c * COLS + col];             // identity residual
            v = fmaxf(v, 0.f);
            y[base + (long)c * (T_ * V_) + col] = v;
        }
    }
}

// ---------------------------------------------------------------------------
extern "C" void kernel_launch(void* const* d_in, const int* in_sizes, int n_in,
                              void* d_out, int out_size, void* d_ws, size_t ws_size,
                              hipStream_t stream)
{
    const float* x     = (const float*)d_in[0];
    const float* PA    = (const float*)d_in[1];
    const float* EI    = (const float*)d_in[2];
    const float* Wq    = (const float*)d_in[3];
    const float* bq    = (const float*)d_in[4];
    const float* Wk    = (const float*)d_in[5];
    const float* bk    = (const float*)d_in[6];
    const float* Ww1   = (const float*)d_in[7];
    const float* bw1   = (const float*)d_in[8];
    const float* Ww2   = (const float*)d_in[9];
    const float* bw2   = (const float*)d_in[10];
    const float* conf  = (const float*)d_in[11];
    const float* alpha = (const float*)d_in[12];
    const float* Wd    = (const float*)d_in[13];
    const float* bd    = (const float*)d_in[14];
    const float* bng   = (const float*)d_in[15];
    const float* bnb   = (const float*)d_in[16];
    const float* bnm   = (const float*)d_in[17];
    const float* bnv   = (const float*)d_in[18];

    float* y      = (float*)d_out;
    float* afused = y + YSIZE;          // second tuple output
    float* tx     = (float*)d_ws;       // N*C*V floats of scratch

    hg_tmean<<<N_ * C_, 256, 0, stream>>>(x, tx);

    const size_t sm2 = (size_t)(1600 + 3200 + 3200 + 3200 + 5000 + 200 + 8 + 8 +
                                625 + 25 + 25) * sizeof(float);
    hg_attn<<<N_, 256, sm2, stream>>>(tx, PA, EI, Wq, bq, Wk, bk, Ww1, bw1,
                                      Ww2, bw2, conf, alpha, afused);

    const size_t sm3 = (size_t)(25600 + 25600 + 4096 + 650 + 25 + 64 + 64 + 64)
                       * sizeof(float);
    hg_fused<<<N_ * (T_ / TCHUNK), 256, sm3, stream>>>(
        x, Wd, bd, bng, bnb, bnm, bnv, afused, y);
}